// KumaMask_4415226380947
// MI455X (gfx1250) — compile-verified
//
#include <hip/hip_runtime.h>
#include <hip/hip_bf16.h>
#include <math.h>

typedef float v2f __attribute__((ext_vector_type(2)));
typedef float v8f __attribute__((ext_vector_type(8)));

#define D_DIM 512
#define CHUNK 64
#define NCHUNKS (D_DIM / CHUNK)          // 8
#define ROWSTRIDE (CHUNK + 4)            // 68 floats: conflict-free K=4 fragment reads
#define WAVES_PER_BLOCK 4
#define ROWS_PER_WAVE 16
#define ROWS_PER_BLOCK (WAVES_PER_BLOCK * ROWS_PER_WAVE)  // 64
#define BLOCK_THREADS (WAVES_PER_BLOCK * 32)              // 128

__device__ __forceinline__ float softplus_f(float x) {
    return fmaxf(x, 0.0f) + log1pf(expf(-fabsf(x)));
}

__device__ __forceinline__ uint32_t rotl32(uint32_t v, int n) {
    return (v << n) | (v >> (32 - n));
}

// JAX threefry2x32 with key = (0, 42)  [jax.random.key(42)]
__device__ __forceinline__ void threefry2x32_42(uint32_t c0, uint32_t c1,
                                                uint32_t& o0, uint32_t& o1) {
    const uint32_t k0 = 0u, k1 = 42u;
    const uint32_t k2 = k0 ^ k1 ^ 0x1BD11BDAu;
    uint32_t ks[3] = {k0, k1, k2};
    uint32_t x0 = c0 + ks[0];
    uint32_t x1 = c1 + ks[1];
    const int rot[2][4] = {{13, 15, 26, 6}, {17, 29, 16, 24}};
#pragma unroll
    for (int i = 0; i < 5; ++i) {
        const int* r = rot[i & 1];
#pragma unroll
        for (int j = 0; j < 4; ++j) {
            x0 += x1;
            x1 = rotl32(x1, r[j]);
            x1 ^= x0;
        }
        x0 += ks[(i + 1) % 3];
        x1 += ks[(i + 2) % 3] + (uint32_t)(i + 1);
    }
    o0 = x0;
    o1 = x1;
}

__global__ __launch_bounds__(BLOCK_THREADS)
void kuma_fused_wmma(const float* __restrict__ x,
                     const float* __restrict__ Wa, const float* __restrict__ ba,
                     const float* __restrict__ Wb, const float* __restrict__ bb,
                     float* __restrict__ out, int rows) {
    __shared__ float sW[2][D_DIM];                                      // Wa | Wb
    __shared__ float sX[WAVES_PER_BLOCK][2][ROWS_PER_WAVE][ROWSTRIDE];  // double-buffered tiles

    const int tid  = threadIdx.x;
    const int lane = tid & 31;
    const int wave = tid >> 5;
    const int n    = lane & 15;   // M-row for A / N-col for B
    const int half = lane >> 4;   // K-half selector within a fragment

    // Stage Wa / Wb into LDS (coalesced across the block); only cross-wave sync needed
    for (int i = tid; i < D_DIM; i += BLOCK_THREADS) {
        sW[0][i] = Wa[i];
        sW[1][i] = Wb[i];
    }
    __syncthreads();

    const int rowBase = blockIdx.x * ROWS_PER_BLOCK + wave * ROWS_PER_WAVE;
    const bool fullTile = (rowBase + ROWS_PER_WAVE) <= rows;  // wave-uniform

    const int loadRow = 2 * 0 + half;  // (placeholder, per-i below)
    (void)loadRow;

    // Issue 8 async global->LDS b128 loads covering one 16x64 tile chunk.
    // lanes 0-15 cover row 2*i (contiguous 256B), lanes 16-31 row 2*i+1.
    auto issue_async_chunk = [&](int buf, int k0) {
        uint32_t ldsBase = (uint32_t)(uintptr_t)(&sX[wave][buf][0][0]);  // LDS byte offset
#pragma unroll
        for (int i = 0; i < 8; ++i) {
            int row = 2 * i + half;
            int col = n * 4;
            uint32_t ldsAddr = ldsBase + (uint32_t)(row * ROWSTRIDE + col) * 4u;
            const float* gp = &x[(size_t)(rowBase + row) * D_DIM + (size_t)(k0 + col)];
            asm volatile("global_load_async_to_lds_b128 %0, %1, off"
                         :: "v"(ldsAddr), "v"(gp)
                         : "memory");
        }
    };

    // Fallback (partial tile): masked VGPR loads + ds stores (same-wave, in-order)
    auto stage_masked_chunk = [&](int buf, int k0) {
#pragma unroll
        for (int i = 0; i < 8; ++i) {
            int row = 2 * i + half;
            int col = n * 4;
            int gr  = rowBase + row;
            float4 v = make_float4(0.f, 0.f, 0.f, 0.f);
            if (gr < rows) {
                v = *reinterpret_cast<const float4*>(
                        &x[(size_t)gr * D_DIM + (size_t)(k0 + col)]);
            }
            sX[wave][buf][row][col + 0] = v.x;
            sX[wave][buf][row][col + 1] = v.y;
            sX[wave][buf][row][col + 2] = v.z;
            sX[wave][buf][row][col + 3] = v.w;
        }
        __builtin_amdgcn_wave_barrier();
    };

    v8f c = {};  // 16x16 f32 accumulator: [N=0]=x.Wa, [N=1]=x.Wb

    if (fullTile) {
        issue_async_chunk(0, 0);  // prime the pipeline
    }

    for (int ci = 0; ci < NCHUNKS; ++ci) {
        const int cur = ci & 1;
        const int k0  = ci * CHUNK;

        if (fullTile) {
            if (ci + 1 < NCHUNKS) {
                // WAR fence: make sure previous fragment reads of buf[1-cur] retired
                asm volatile("s_wait_dscnt 0x0" ::: "memory");
                issue_async_chunk(1 - cur, k0 + CHUNK);
                // async loads complete in order: <=8 outstanding => chunk ci landed
                asm volatile("s_wait_asynccnt 0x8" ::: "memory");
            } else {
                asm volatile("s_wait_asynccnt 0x0" ::: "memory");
            }
        } else {
            stage_masked_chunk(cur, k0);
        }

        // ---- 16 x V_WMMA_F32_16X16X4_F32 over this chunk ----
#pragma unroll
        for (int kk = 0; kk < CHUNK; kk += 4) {
            // A fragment (ISA 7.12.2): lane holds row M=n; half 0 -> K=kk,kk+1, half 1 -> K=kk+2,kk+3
            v2f af;
            af.x = sX[wave][cur][n][kk + 2 * half + 0];
            af.y = sX[wave][cur][n][kk + 2 * half + 1];
            // B fragment: column N=n; only N=0 (Wa) and N=1 (Wb) are non-zero
            v2f bf = {0.0f, 0.0f};
            if (n < 2) {
                const float* ws = sW[n];
                bf.x = ws[k0 + kk + 2 * half + 0];
                bf.y = ws[k0 + kk + 2 * half + 1];
            }
            c = __builtin_amdgcn_wmma_f32_16x16x4_f32(
                    false, af, false, bf, (short)0, c, false, false);
        }
        __builtin_amdgcn_wave_barrier();
    }

    // ---- spill C to LDS: VGPR v, lanes 0-15 -> (M=v, N=n); lanes 16-31 -> (M=v+8, N=n) ----
#pragma unroll
    for (int v = 0; v < 8; ++v) {
        sX[wave][0][v + 8 * half][n] = c[v];
    }
    __builtin_amdgcn_wave_barrier();  // same-wave LDS ops are in-order; fence the scheduler

    // ---- fused Kumaraswamy epilogue: one row per lane (lanes 0-15) ----
    if (half == 0) {
        int r = rowBase + n;
        if (r < rows) {
            float adot = sX[wave][0][n][0];
            float bdot = sX[wave][0][n][1];
            float a = softplus_f(adot + ba[0]);
            float b = softplus_f(bdot + bb[0]);

            // jax.random.uniform(key(42), (rows,)): iota counters split in halves
            uint32_t halfn = (uint32_t)rows >> 1;
            uint32_t ur = (uint32_t)r;
            uint32_t c0, c1, o0, o1;
            if (ur < halfn) { c0 = ur;          c1 = ur + halfn; }
            else            { c0 = ur - halfn;  c1 = ur;         }
            threefry2x32_42(c0, c1, o0, o1);
            uint32_t bits = (ur < halfn) ? o0 : o1;

            const float tiny = 1.17549435e-38f;
            float u01 = __uint_as_float((bits >> 9) | 0x3F800000u) - 1.0f;  // [0,1)
            float u = fmaxf(tiny, u01 * (1.0f - tiny) + tiny);

            float t   = powf(u, 1.0f / b);
            float kum = powf(1.0f - t, 1.0f / a);
            float z   = kum * 1.2f - 0.1f;   // stretch to [-0.1, 1.1]
            z = fminf(fmaxf(z, 0.0f), 1.0f); // rectify
            out[r] = z;
        }
    }
}

extern "C" void kernel_launch(void* const* d_in, const int* in_sizes, int n_in,
                              void* d_out, int out_size, void* d_ws, size_t ws_size,
                              hipStream_t stream) {
    (void)n_in; (void)d_ws; (void)ws_size; (void)in_sizes;
    const float* x  = (const float*)d_in[0];
    const float* Wa = (const float*)d_in[1];
    const float* ba = (const float*)d_in[2];
    const float* Wb = (const float*)d_in[3];
    const float* bb = (const float*)d_in[4];
    float* out = (float*)d_out;

    int rows = out_size;  // B*T = 131072
    int blocks = (rows + ROWS_PER_BLOCK - 1) / ROWS_PER_BLOCK;
    kuma_fused_wmma<<<blocks, BLOCK_THREADS, 0, stream>>>(x, Wa, ba, Wb, bb, out, rows);
}